// MRConv1d_6296422056171
// MI455X (gfx1250) — compile-verified
//
#include <hip/hip_runtime.h>

#define B_    4
#define C_    128
#define N_    10000
#define K_    16
#define OUT_  128
#define TWO_C 256
#define TILES_PER_BLK 5                 // 5 x 16 = 80 points per block; 10000 = 80*125
#define CHUNK (16 * TILES_PER_BLK)
#define NEG_BIG (-3.402823466e38f)

typedef float v2f __attribute__((ext_vector_type(2)));
typedef float v4f __attribute__((ext_vector_type(4)));
typedef float v8f __attribute__((ext_vector_type(8)));

// ---------------------------------------------------------------------------
// Kernel 1: transpose x [B,C,N] -> xT [B,N,C] so each neighbor gather becomes
// one contiguous 512B row read (C=128 floats) instead of 128 strided 4B reads.
// ---------------------------------------------------------------------------
__global__ __launch_bounds__(256) void transpose_x_kernel(
    const float* __restrict__ x, float* __restrict__ xT) {
  __shared__ float tile[32][33];  // +1 pad: conflict-free transpose
  const int b  = blockIdx.z;
  const int c0 = blockIdx.y * 32;
  const int n0 = blockIdx.x * 32;
  const int tx = threadIdx.x;     // 0..31
  const int ty = threadIdx.y;     // 0..7
  const float* xb  = x  + (size_t)b * C_ * N_;
  float*       xTb = xT + (size_t)b * N_ * C_;
#pragma unroll
  for (int i = 0; i < 4; ++i) {
    const int c = c0 + ty + i * 8;
    const int n = n0 + tx;
    float v = 0.f;
    if (n < N_) v = xb[(size_t)c * N_ + n];   // coalesced along n
    tile[ty + i * 8][tx] = v;                 // tile[c_local][n_local]
  }
  __syncthreads();
#pragma unroll
  for (int i = 0; i < 4; ++i) {
    const int n = n0 + ty + i * 8;
    const int c = c0 + tx;
    if (n < N_) xTb[(size_t)n * C_ + c] = tile[tx][ty + i * 8];  // coalesced along c
  }
}

// ---------------------------------------------------------------------------
// Kernel 2 (fused): block = (b, 80-wide n-chunk). Each of 8 waves owns one
// 16-row o-tile and PRELOADS its entire A-matrix slice of W into 128 VGPRs
// (64 x v2f WMMA fragments), then loops over 5 n-tiles:
//   stage indices + x tile in LDS -> max-relative gather rows -> WMMA K-loop
//   (pure ds_load + v_wmma, zero global traffic) -> bias+ReLU store.
// ---------------------------------------------------------------------------
__global__ __launch_bounds__(256) void mrconv_fused_kernel(
    const float* __restrict__ x, const int* __restrict__ eidx,
    const float* __restrict__ W, const float* __restrict__ bias,
    const float* __restrict__ xT, int useT, float* __restrict__ out) {
  __shared__ float h[TWO_C][17];   // rows 0..127: x tile; 128..255: x_max tile
  __shared__ int idxJ[16][K_];     // edge_index[0] (neighbor)
  __shared__ int idxI[16][K_];     // edge_index[1] (center)

  const int b     = blockIdx.x / (N_ / CHUNK);
  const int nbase = (blockIdx.x % (N_ / CHUNK)) * CHUNK;
  const int tid   = threadIdx.x;               // 0..255
  const int lane  = tid & 31;
  const int wave  = tid >> 5;                  // 0..7

  const float* xb  = x  + (size_t)b * C_ * N_;
  const float* xTb = xT + (size_t)b * N_ * C_;

  // ---- per-wave A-fragment preload: W[o0:o0+16, 0:256] in 128 VGPRs ------
  // A 16x4 f32 layout: lane L -> M = L%16, K = 2*(L/16) + {vgpr0,vgpr1}
  const int o0 = wave * 16;
  const int mc = lane & 15;          // A row M == B/D column N
  const int kk = (lane >> 4) * 2;    // K sub-offset
  v2f af[TWO_C / 4];
#pragma unroll
  for (int i = 0; i < TWO_C / 4; ++i) {
    af[i] = *(const v2f*)(W + (size_t)(o0 + mc) * TWO_C + i * 4 + kk);
  }
  const float bb0 = bias[o0 + (lane >> 4) * 8 + 0];  // cache bias per lane rows
  // (bias re-read per row in epilogue is cheap/L0-resident; keep simple)
  (void)bb0;

  for (int t = 0; t < TILES_PER_BLK; ++t) {
    const int n0 = nbase + t * 16;
    if (t) __syncthreads();  // all waves done reading h of previous tile

    // ---- stage edge indices (16 n x 16 k x 2 ints) -----------------------
    for (int i = tid; i < 16 * K_; i += 256) {
      const int n = i / K_, k = i % K_;
      idxJ[n][k] = eidx[(((size_t)0 * B_ + b) * N_ + (n0 + n)) * K_ + k];
      idxI[n][k] = eidx[(((size_t)1 * B_ + b) * N_ + (n0 + n)) * K_ + k];
    }
    // ---- stage x tile: h[c][nn] = x[b,c,n0+nn] ---------------------------
    for (int i = tid; i < C_ * 16; i += 256) {
      const int c = i >> 4, nn = i & 15;
      h[c][nn] = xb[(size_t)c * N_ + (n0 + nn)];
    }
    __syncthreads();

    // ---- max-relative: wave w handles n = 2w, 2w+1; lane owns 4 channels -
#pragma unroll
    for (int p = 0; p < 2; ++p) {
      const int n = wave * 2 + p;
      float m0 = NEG_BIG, m1 = NEG_BIG, m2 = NEG_BIG, m3 = NEG_BIG;
      for (int k = 0; k < K_; ++k) {
        const int j = idxJ[n][k];   // wave-uniform -> LDS broadcast
        const int i = idxI[n][k];
        float j0, j1, j2, j3, i0, i1, i2, i3;
        if (useT) {
          // one fully-coalesced 512B row per wave (lane*16B, contiguous)
          v4f vj = *((const v4f*)(xTb + (size_t)j * C_) + lane);
          v4f vi = *((const v4f*)(xTb + (size_t)i * C_) + lane);
          j0 = vj.x; j1 = vj.y; j2 = vj.z; j3 = vj.w;
          i0 = vi.x; i1 = vi.y; i2 = vi.z; i3 = vi.w;
        } else {
          const int c = lane * 4;   // fallback: strided channel reads
          j0 = xb[(size_t)(c + 0) * N_ + j]; i0 = xb[(size_t)(c + 0) * N_ + i];
          j1 = xb[(size_t)(c + 1) * N_ + j]; i1 = xb[(size_t)(c + 1) * N_ + i];
          j2 = xb[(size_t)(c + 2) * N_ + j]; i2 = xb[(size_t)(c + 2) * N_ + i];
          j3 = xb[(size_t)(c + 3) * N_ + j]; i3 = xb[(size_t)(c + 3) * N_ + i];
        }
        m0 = fmaxf(m0, j0 - i0);
        m1 = fmaxf(m1, j1 - i1);
        m2 = fmaxf(m2, j2 - i2);
        m3 = fmaxf(m3, j3 - i3);
      }
      const int c = C_ + lane * 4;
      h[c + 0][n] = m0; h[c + 1][n] = m1; h[c + 2][n] = m2; h[c + 3][n] = m3;
    }
    __syncthreads();

    // ---- WMMA K-loop: D(16x16) += A(16x4) x B(4x16), 64 steps ------------
    // B 4x16 f32 layout: lane L -> N = L%16, K = 2*(L/16) + {vgpr0,vgpr1}
    v8f acc = {0.f, 0.f, 0.f, 0.f, 0.f, 0.f, 0.f, 0.f};
#pragma unroll
    for (int i = 0; i < TWO_C / 4; ++i) {
      v2f bb;
      bb.x = h[i * 4 + kk + 0][mc];
      bb.y = h[i * 4 + kk + 1][mc];
      acc = __builtin_amdgcn_wmma_f32_16x16x4_f32(
          /*neg_a=*/false, af[i], /*neg_b=*/false, bb,
          /*c_mod=*/(short)0, acc, /*reuse_a=*/false, /*reuse_b=*/false);
    }

    // ---- epilogue: C/D layout -> VGPR r: M = r + 8*(lane/16), N = lane%16
    const int mhalf = (lane >> 4) * 8;
#pragma unroll
    for (int r = 0; r < 8; ++r) {
      const int o = o0 + r + mhalf;
      float v = acc[r] + bias[o];
      v = fmaxf(v, 0.f);
      out[((size_t)b * OUT_ + o) * N_ + (n0 + mc)] = v;
    }
  }
}

// ---------------------------------------------------------------------------
extern "C" void kernel_launch(void* const* d_in, const int* in_sizes, int n_in,
                              void* d_out, int out_size, void* d_ws, size_t ws_size,
                              hipStream_t stream) {
  const float* x    = (const float*)d_in[0];
  const int*   eidx = (const int*)d_in[1];
  const float* W    = (const float*)d_in[2];
  const float* bias = (const float*)d_in[3];
  float*       out  = (float*)d_out;

  const size_t xT_bytes = (size_t)B_ * N_ * C_ * sizeof(float);
  const int useT = (ws_size >= xT_bytes) ? 1 : 0;
  float* xT = useT ? (float*)d_ws : (float*)x;  // pointer unused when !useT

  if (useT) {
    dim3 tb(32, 8);
    dim3 tg((N_ + 31) / 32, C_ / 32, B_);
    transpose_x_kernel<<<tg, tb, 0, stream>>>(x, xT);
  }
  mrconv_fused_kernel<<<B_ * (N_ / CHUNK), 256, 0, stream>>>(
      x, eidx, W, bias, xT, useT, out);
}